// ConvEncoder_10849087390293
// MI455X (gfx1250) — compile-verified
//
#include <hip/hip_runtime.h>

// ConvCNP set->grid encoder for MI455X (gfx1250, wave32).
//   A=4 batches, N=1024 context, M=8640 grid, C=64 channels.
// Phase 1 (VALU/trans bound): per-grid-point RBF reduction over context (LDS-staged).
// Phase 2 (matrix): feats(16x2 zero-padded to 16x4) @ w(4x16) + b via V_WMMA_F32_16X16X4_F32.

typedef float v2f __attribute__((ext_vector_type(2)));
typedef float v8f __attribute__((ext_vector_type(8)));

#define A_BATCH 4
#define NCTX    1024
#define MGRID   8640
#define COUT    64
#define THREADS 96          // 3 wave32; 96 grid points per block; 8640 = 90 * 96

__global__ __launch_bounds__(THREADS)
void convcnp_encoder_kernel(const float* __restrict__ x_ctx,
                            const float* __restrict__ y_ctx,
                            const float* __restrict__ x_grid,
                            const float* __restrict__ sigma,
                            const float* __restrict__ wmat,   // (2, 64) row-major
                            const float* __restrict__ bias,   // (64,)
                            float* __restrict__ out)          // (A, C, M)
{
    __shared__ float ldsX[NCTX];          // context x, pre-scaled by 1/scale
    __shared__ float ldsY[NCTX];          // context y
    __shared__ float ldsD[THREADS];       // density per grid point
    __shared__ float ldsC[THREADS];       // normalized conv per grid point

    const int a      = blockIdx.y;
    const int mblock = blockIdx.x * THREADS;
    const int tid    = threadIdx.x;

    const float inv_scale = __expf(-sigma[0]);   // 1 / exp(sigma)

    // ---- stage context into LDS (scaled so the inner loop is sub+mul+exp) ----
    for (int i = tid; i < NCTX; i += THREADS) {
        ldsX[i] = x_ctx[a * NCTX + i] * inv_scale;
        ldsY[i] = y_ctx[a * NCTX + i];
    }
    __syncthreads();

    // ---- phase 1: RBF reduction over all context points (exp-bound) ----
    const int   m  = mblock + tid;
    const float gx = x_grid[m] * inv_scale;
    const float kf = -0.72134752044448170f;      // -0.5 * log2(e):  exp(-d^2/2) = exp2(kf*d^2)
    float dens = 0.0f, conv = 0.0f;
    #pragma unroll 8
    for (int n = 0; n < NCTX; ++n) {
        float d = gx - ldsX[n];
        float r = __builtin_amdgcn_exp2f(d * d * kf);   // v_exp_f32
        dens += r;
        conv  = fmaf(r, ldsY[n], conv);
    }
    ldsD[tid] = dens;
    ldsC[tid] = conv / (dens + 1e-8f);           // density-normalize
    __syncthreads();

    // ---- phase 2: relu(feats @ w + b) with V_WMMA_F32_16X16X4_F32 ----
    const int wave = tid >> 5;
    const int lane = tid & 31;
    const int hi   = lane >> 4;      // lanes 16-31 carry K=2/3 (zero pad) for A/B operands
    const int ln   = lane & 15;

    // B operand (4x16, K x channels): VGPR0 = rows K=0 / K=2, VGPR1 = rows K=1 / K=3.
    // Only K=0 (w row 0 -> density) and K=1 (w row 1 -> conv) are non-zero.
    v2f bop[4];
    #pragma unroll
    for (int g = 0; g < 4; ++g) {
        const int c = 16 * g + ln;
        bop[g].x = hi ? 0.0f : wmat[0 * COUT + c];
        bop[g].y = hi ? 0.0f : wmat[1 * COUT + c];
    }

    const int p0 = wave * 32;        // each wave owns 32 grid points = 2 A-tiles of 16
    #pragma unroll
    for (int t = 0; t < 2; ++t) {
        // A operand (16x4): lanes 0-15 -> (K=0,K=1) = (density, conv); lanes 16-31 -> zeros.
        v2f aop;
        aop.x = hi ? 0.0f : ldsD[p0 + 16 * t + ln];
        aop.y = hi ? 0.0f : ldsC[p0 + 16 * t + ln];

        #pragma unroll
        for (int g = 0; g < 4; ++g) {
            const int c  = 16 * g + ln;     // output channel for this lane (both halves)
            const float bv = bias[c];
            v8f acc = {bv, bv, bv, bv, bv, bv, bv, bv};   // C pre-loaded with bias

            acc = __builtin_amdgcn_wmma_f32_16x16x4_f32(
                      /*neg_a=*/false, aop, /*neg_b=*/false, bop[g],
                      /*c_mod=*/(short)0, acc, /*reuse_a=*/false, /*reuse_b=*/false);

            // D layout: VGPR r holds rows (r, r+8); this lane's 8 results are
            // 8 consecutive m values for channel c -> two aligned b128 stores.
            float v0 = fmaxf(acc[0], 0.0f), v1 = fmaxf(acc[1], 0.0f);
            float v2 = fmaxf(acc[2], 0.0f), v3 = fmaxf(acc[3], 0.0f);
            float v4 = fmaxf(acc[4], 0.0f), v5 = fmaxf(acc[5], 0.0f);
            float v6 = fmaxf(acc[6], 0.0f), v7 = fmaxf(acc[7], 0.0f);

            float* dst = out + ((size_t)a * COUT + c) * MGRID
                             + (mblock + p0 + 16 * t + 8 * hi);
            *reinterpret_cast<float4*>(dst)     = make_float4(v0, v1, v2, v3);
            *reinterpret_cast<float4*>(dst + 4) = make_float4(v4, v5, v6, v7);
        }
    }
}

extern "C" void kernel_launch(void* const* d_in, const int* in_sizes, int n_in,
                              void* d_out, int out_size, void* d_ws, size_t ws_size,
                              hipStream_t stream) {
    const float* x_ctx  = (const float*)d_in[0];   // (A, N, 1)
    const float* y_ctx  = (const float*)d_in[1];   // (A, N, 1)
    const float* x_grid = (const float*)d_in[2];   // (M, 1)
    const float* sigma  = (const float*)d_in[3];   // (1,)
    const float* wmat   = (const float*)d_in[4];   // (2, C)
    const float* bias   = (const float*)d_in[5];   // (C,)
    float* out = (float*)d_out;                    // (A, C, M)

    dim3 grid(MGRID / THREADS, A_BATCH);           // 90 x 4 blocks, no tail
    convcnp_encoder_kernel<<<grid, THREADS, 0, stream>>>(
        x_ctx, y_ctx, x_grid, sigma, wmat, bias, out);
}